// MaxSigmoidAttnBlock_26774826123900
// MI455X (gfx1250) — compile-verified
//
#include <hip/hip_runtime.h>
#include <hip/hip_bf16.h>

typedef float v2f __attribute__((ext_vector_type(2)));
typedef float v8f __attribute__((ext_vector_type(8)));

#define B_    8
#define C_    128
#define H_    64
#define W_    64
#define HW_   4096
#define GC_   512
#define QEC_  64
#define KTOT_ 4608   // 9 taps * 512 ic

// LDS tile geometry for main kernel (k-pair interleaved):
//   sB[kpair 0..31][row 0..2][px 0..65][2]
#define RS2_  136           // row stride in floats (66 px-pairs padded to 68)
#define KPS2_ 416           // kpair stride in floats (3*136=408 padded; 416 % 64 == 32)
#define NSTAGE_ 12672       // 64*3*66 elements staged per chunk

// Hamilton product expansion: Wbig[q][comp] = qsgn * w_component[qsel]
__constant__ int   qsel[16] = {0,1,2,3,  1,0,3,2,  2,3,0,1,  3,2,1,0};
__constant__ float qsgn[16] = {1,-1,-1,-1,  1,1,-1,1,  1,1,1,-1,  1,-1,1,1};

// ---- g[b, n*64+m*16+c] = guide @ wg.T + bg  (8 x 256) ----
__global__ void k_guide_gemm(const float* __restrict__ guide, const float* __restrict__ wg,
                             const float* __restrict__ bg, float* __restrict__ g_ws) {
    int t = blockIdx.x * blockDim.x + threadIdx.x;   // 2048
    int j = t & 255, b = t >> 8;
    const float* gu = guide + b * GC_;
    const float* wr = wg + (size_t)j * GC_;
    float s = bg[j];
    for (int k = 0; k < GC_; ++k) s += gu[k] * wr[k];
    g_ws[b * 256 + j] = s;
}

// ---- A_comb[b, (m*4+n), ic] = sum_{q,c} sgn(q,comp)*w_ec[sel][n*16+c][cin]*g[b,n,m,c] ----
__global__ void k_acomb(const float* __restrict__ w_ec, const float* __restrict__ g_ws,
                        float* __restrict__ acomb) {
    int t = blockIdx.x * blockDim.x + threadIdx.x;   // 65536
    int ic = t & 511;
    int mn = (t >> 9) & 15;
    int b  = t >> 13;
    int m = mn >> 2, n = mn & 3;
    int comp = ic >> 7, cin = ic & 127;
    const float* g = g_ws + b * 256 + n * 64 + m * 16;
    float s = 0.f;
    for (int q = 0; q < 4; ++q) {
        int   sl = qsel[q * 4 + comp];
        float sg = qsgn[q * 4 + comp];
        const float* wp = w_ec + (size_t)(sl * QEC_ + n * 16) * C_ + cin;
        float ps = 0.f;
        for (int c = 0; c < 16; ++c) ps += wp[c * C_] * g[c];
        s += sg * ps;
    }
    acomb[(size_t)b * 8192 + mn * 512 + ic] = s;
}

// ---- awb[b, m*4+n] = sum_{q,c} b_ec[q][n*16+c] * g[b,n,m,c]  (embed bias fold) ----
__global__ void k_awbias(const float* __restrict__ b_ec, const float* __restrict__ g_ws,
                         float* __restrict__ awb) {
    int t = threadIdx.x;                              // 128
    int mn = t & 15, b = t >> 4;
    int m = mn >> 2, n = mn & 3;
    const float* g = g_ws + b * 256 + n * 64 + m * 16;
    float s = 0.f;
    for (int q = 0; q < 4; ++q) {
        const float* bp = b_ec + q * 64 + n * 16;
        for (int c = 0; c < 16; ++c) s += bp[c] * g[c];
    }
    awb[b * 16 + mn] = s;
}

// ---- Wexp[oc][tap*512+ic] = sgn(q,comp) * w_proj[sel][o][cin][tap]  (quaternion expand) ----
__global__ void k_wexpand(const float* __restrict__ w_proj, float* __restrict__ wexp) {
    int t = blockIdx.x * blockDim.x + threadIdx.x;   // 256*4608
    int k  = t % KTOT_;
    int oc = t / KTOT_;
    int tap = k >> 9;
    int ic  = k & 511;
    int comp = ic >> 7, cin = ic & 127;
    int q = oc >> 6, o = oc & 63;
    int   sl = qsel[q * 4 + comp];
    float sg = qsgn[q * 4 + comp];
    wexp[t] = sg * w_proj[(size_t)((sl * 64 + o) * 128 + cin) * 9 + tap];
}

// ---- attention map: aw[b,m,h,w] = sigmoid(max_n(Acomb . xin)/4 + bias[m]) * scale[m] ----
__global__ __launch_bounds__(256) void k_attn(const float* __restrict__ x,
        const float* __restrict__ acomb, const float* __restrict__ awb,
        const float* __restrict__ attn_bias, const float* __restrict__ scale_p,
        float* __restrict__ aw_ws) {
    __shared__ float sA[16 * 512];
    int tid = threadIdx.x;
    int pix = blockIdx.x * 256 + tid;                // 32768 pixels; whole block same batch
    int b   = pix >> 12;
    int off = pix & 4095;
    for (int i = tid; i < 8192; i += 256) sA[i] = acomb[(size_t)b * 8192 + i];
    __syncthreads();
    float acc[16];
#pragma unroll
    for (int j = 0; j < 16; ++j) acc[j] = awb[b * 16 + j];
    const float* xb = x + (size_t)b * C_ * 4 * HW_ + off;
    for (int ic = 0; ic < 512; ++ic) {
        int comp = ic >> 7, cin = ic & 127;
        float xv = xb[(size_t)(cin * 4 + comp) * HW_];
#pragma unroll
        for (int j = 0; j < 16; ++j) acc[j] += sA[j * 512 + ic] * xv;
    }
#pragma unroll
    for (int m = 0; m < 4; ++m) {
        float s = fmaxf(fmaxf(acc[m * 4 + 0], acc[m * 4 + 1]),
                        fmaxf(acc[m * 4 + 2], acc[m * 4 + 3]));
        float z   = s * 0.25f + attn_bias[m];
        float sig = 1.f / (1.f + __expf(-z));
        aw_ws[((b * 4 + m) << 12) + off] = sig * scale_p[m];
    }
}

// ---- main: 3x3 quaternion conv as implicit GEMM via V_WMMA_F32_16X16X4_F32 ----
// One workgroup = one (b,h) output row (N=64 pixels -> 4 tiles per wave).
// 16 waves/block; wave wv -> oc rows [wv*16, wv*16+16).
// Per ic-chunk: stage 64ic x 3rows x 66px halo once (k-pair interleaved),
// run all 9 taps from LDS; B operands are single conflict-free ds_load_b64.
__global__ __launch_bounds__(512) void k_qconv3x3_wmma(const float* __restrict__ x,
        const float* __restrict__ wexp, const float* __restrict__ b_proj,
        const float* __restrict__ aw_ws, float* __restrict__ out) {
    __shared__ float sB[32 * KPS2_];                 // 52 KB
    const int tid  = threadIdx.x;
    const int lane = tid & 31;
    const int wv   = tid >> 5;                       // 0..15
    const int b    = blockIdx.x >> 6;                // grid = B*H = 512
    const int h    = blockIdx.x & 63;

    const int mrow = lane & 15;                      // A/D row within tile
    const int kb   = (lane >> 4) << 1;               // K pair base {0,2}
    const int oc0  = wv << 4;

    const float* xb   = x + (size_t)b * C_ * 4 * HW_;
    const float* aRow = wexp + (size_t)(oc0 + mrow) * KTOT_ + kb;

    v8f acc0 = {0.f,0.f,0.f,0.f,0.f,0.f,0.f,0.f};
    v8f acc1 = acc0, acc2 = acc0, acc3 = acc0;

    for (int icc = 0; icc < 8; ++icc) {
        const int ic0 = icc << 6;
        // ---- stage 64 ic x 3 rows x 66 px (zero-padded halo), k-pair interleaved ----
        for (int e = tid; e < NSTAGE_; e += 512) {
            int icl  = e / 198;                      // 198 = 3*66
            int rem2 = e - icl * 198;
            int r    = rem2 / 66;
            int px   = rem2 - r * 66;
            int ic   = ic0 + icl;
            int comp = ic >> 7, cin = ic & 127;
            int hh   = h + r - 1;
            int ww   = px - 1;
            float v = 0.f;
            if ((unsigned)hh < (unsigned)H_ && (unsigned)ww < (unsigned)W_)
                v = xb[(size_t)(cin * 4 + comp) * HW_ + hh * W_ + ww];
            sB[(icl >> 1) * KPS2_ + r * RS2_ + px * 2 + (icl & 1)] = v;
        }
        __syncthreads();

        // ---- 9 taps x 16 K-steps x 4 pixel tiles ----
        for (int tap = 0; tap < 9; ++tap) {
            const int kh = tap / 3, kw = tap - kh * 3;
            const float* ap = aRow + tap * 512 + ic0;
            v2f areg[16];
#pragma unroll
            for (int s = 0; s < 16; ++s) areg[s] = *(const v2f*)(ap + s * 4);
            // one base VGPR; all per-(s,t) deltas become 16-bit DS immediates
            const float* bbase = sB + (lane >> 4) * KPS2_ + kh * RS2_
                                    + (kw + (lane & 15)) * 2;
#pragma unroll
            for (int s = 0; s < 16; ++s) {
                const float* bp = bbase + s * (2 * KPS2_);
                v2f bv0 = *(const v2f*)(bp +  0);
                v2f bv1 = *(const v2f*)(bp + 32);
                v2f bv2 = *(const v2f*)(bp + 64);
                v2f bv3 = *(const v2f*)(bp + 96);
                acc0 = __builtin_amdgcn_wmma_f32_16x16x4_f32(false, areg[s], false, bv0, (short)0, acc0, false, false);
                acc1 = __builtin_amdgcn_wmma_f32_16x16x4_f32(false, areg[s], false, bv1, (short)0, acc1, false, false);
                acc2 = __builtin_amdgcn_wmma_f32_16x16x4_f32(false, areg[s], false, bv2, (short)0, acc2, false, false);
                acc3 = __builtin_amdgcn_wmma_f32_16x16x4_f32(false, areg[s], false, bv3, (short)0, acc3, false, false);
            }
        }
        __syncthreads();
    }

    // ---- epilogue: + b_proj, * aw, store in [b, o, q, h, w] order ----
    const int q    = wv >> 2;
    const int nh   = wv & 3;
    const int o0   = nh << 4;
    const int n    = lane & 15;
    const int half = lane >> 4;
    const float* awrow = aw_ws + (((b << 2) + nh) << 12) + h * W_;
    v8f accs[4] = {acc0, acc1, acc2, acc3};
#pragma unroll
    for (int t = 0; t < 4; ++t) {
        const float awv = awrow[t * 16 + n];
#pragma unroll
        for (int v = 0; v < 8; ++v) {
            const int mr = v + (half << 3);
            const int o  = o0 + mr;
            float val = (accs[t][v] + b_proj[q * 64 + o]) * awv;
            out[(((size_t)(b * 64 + o) * 4 + q) << 12) + h * W_ + t * 16 + n] = val;
        }
    }
}

extern "C" void kernel_launch(void* const* d_in, const int* in_sizes, int n_in,
                              void* d_out, int out_size, void* d_ws, size_t ws_size,
                              hipStream_t stream) {
    const float* x      = (const float*)d_in[0];
    const float* guide  = (const float*)d_in[1];
    const float* w_ec   = (const float*)d_in[2];
    const float* b_ec   = (const float*)d_in[3];
    const float* wg     = (const float*)d_in[4];
    const float* bg     = (const float*)d_in[5];
    const float* attn_b = (const float*)d_in[6];
    const float* scalep = (const float*)d_in[7];
    const float* w_proj = (const float*)d_in[8];
    const float* b_proj = (const float*)d_in[9];
    float* out = (float*)d_out;
    float* ws  = (float*)d_ws;

    float* g_ws   = ws;                // 2048 floats
    float* awb_ws = ws + 2048;         // 128
    float* acomb  = ws + 4096;         // 65536
    float* aw_ws  = ws + 69632;        // 131072
    float* wexp   = ws + 200704;       // 1179648  (total ~5.5 MB)

    k_guide_gemm<<<8,    256, 0, stream>>>(guide, wg, bg, g_ws);
    k_acomb     <<<256,  256, 0, stream>>>(w_ec, g_ws, acomb);
    k_awbias    <<<1,    128, 0, stream>>>(b_ec, g_ws, awb_ws);
    k_wexpand   <<<4608, 256, 0, stream>>>(w_proj, wexp);
    k_attn      <<<128,  256, 0, stream>>>(x, acomb, awb_ws, attn_b, scalep, aw_ws);
    k_qconv3x3_wmma<<<512, 512, 0, stream>>>(x, wexp, b_proj, aw_ws, out);
}